// CustomLoss_14010183319734
// MI455X (gfx1250) — compile-verified
//
#include <hip/hip_runtime.h>
#include <cmath>

#define N_PRED 76800          // 3*160*160 predictions per image
#define NDIM   9
#define NT     32             // targets per image
#define NB     16             // batch
#define BLK    256            // threads per block (8 waves of 32)
#define CPB    (N_PRED / BLK) // 300 chunks per image
#define NCH    (NB * CPB)     // 4800 chunk-blocks total

typedef float v2f __attribute__((ext_vector_type(2)));
typedef float v8f __attribute__((ext_vector_type(8)));

// ---------------- feature probes (device pass only) ----------------
#if defined(__AMDGCN__)
 #if __has_builtin(__builtin_amdgcn_global_load_async_to_lds_b32)
  #define HAVE_ASYNC_LDS 1
 #else
  #define HAVE_ASYNC_LDS 0
  #warning "gfx1250 probe: NO __builtin_amdgcn_global_load_async_to_lds_b32"
 #endif
 #if __has_builtin(__builtin_amdgcn_s_wait_asynccnt)
  #define HAVE_WAIT_ASYNC 1
 #else
  #define HAVE_WAIT_ASYNC 0
  #warning "gfx1250 probe: NO __builtin_amdgcn_s_wait_asynccnt"
 #endif
 #if __has_builtin(__builtin_amdgcn_wmma_f32_16x16x4_f32)
  #define HAVE_WMMA 1
 #else
  #define HAVE_WMMA 0
  #warning "gfx1250 probe: NO __builtin_amdgcn_wmma_f32_16x16x4_f32"
 #endif
 #if __has_builtin(__builtin_amdgcn_ballot_w32)
  #define HAVE_BALLOT 1
 #else
  #define HAVE_BALLOT 0
  #warning "gfx1250 probe: NO __builtin_amdgcn_ballot_w32"
 #endif
#else
 #define HAVE_ASYNC_LDS 0
 #define HAVE_WAIT_ASYNC 0
 #define HAVE_WMMA 0
 #define HAVE_BALLOT 0
#endif

#if HAVE_ASYNC_LDS
typedef __attribute__((address_space(1))) int* as1_int_ptr;
typedef __attribute__((address_space(3))) int* as3_int_ptr;
#endif

__device__ __forceinline__ unsigned ballot32(bool p) {
#if HAVE_BALLOT
  return __builtin_amdgcn_ballot_w32(p);
#else
  return (unsigned)__ballot(p);
#endif
}

__device__ __forceinline__ float softplusf(float x) {
  // log(1+exp(x)) computed stably
  return fmaxf(x, 0.f) + log1pf(expf(-fabsf(x)));
}

// Sum 256 floats staged in LDS using V_WMMA_F32_16X16X4_F32 with an all-ones
// B matrix (layout independent) and the documented 16x4 A / 16x16 D layouts.
// Must be called by all 32 lanes of wave 0 (EXEC all ones). Result valid in
// every lane.
__device__ __forceinline__ float reduce256_wave0(const float* buf, int lane) {
#if HAVE_WMMA
  v8f d = {0.f, 0.f, 0.f, 0.f, 0.f, 0.f, 0.f, 0.f};
  v2f bones; bones.x = 1.f; bones.y = 1.f;
  #pragma unroll
  for (int j = 0; j < 4; ++j) {
    const int base = 64 * j;
    // A[m][k] (16x4): value id v = 4*m + k.
    // lanes 0-15  : M=lane,    vgpr0=K0, vgpr1=K1
    // lanes 16-31 : M=lane-16, vgpr0=K2, vgpr1=K3
    v2f a;
    if (lane < 16) {
      a.x = buf[base + 4 * lane + 0];
      a.y = buf[base + 4 * lane + 1];
    } else {
      a.x = buf[base + 4 * (lane - 16) + 2];
      a.y = buf[base + 4 * (lane - 16) + 3];
    }
    // D[m][n] = sum_k A[m][k]*1 + C ; chain C accumulation over 4 tiles
    d = __builtin_amdgcn_wmma_f32_16x16x4_f32(false, a, false, bones,
                                              (short)0, d, false, false);
  }
  // lanes 0-15 hold rows 0..7 in d[0..7]; lanes 16-31 hold rows 8..15
  float s = d[0] + d[1] + d[2] + d[3] + d[4] + d[5] + d[6] + d[7];
  float other = __shfl_xor(s, 16, 32);
  return s + other;
#else
  float s = 0.f;
  for (int i = 0; i < 256; ++i) s += buf[i]; // uniform fallback
  return s;
#endif
}

// ------------------------------------------------------------------
// K1: per chunk of 256 predictions: best-IoU pos/neg classification
// (division-free), ballot masks, chunk pos/neg counts, obj partials.
// ------------------------------------------------------------------
__global__ __launch_bounds__(BLK) void k1_iou(const float* __restrict__ pred,
                                              const float* __restrict__ tgt,
                                              int* __restrict__ chunkPos,
                                              int* __restrict__ chunkNeg,
                                              unsigned* __restrict__ posMask,
                                              float* __restrict__ objPart) {
  const int gid  = blockIdx.x;
  const int img  = gid / CPB;
  const int chunk = gid - img * CPB;
  const int tid  = threadIdx.x;
  const int lane = tid & 31;
  const int w    = tid >> 5;

  __shared__ float st1x[NT], st1y[NT], st2x[NT], st2y[NT], sta[NT];
  __shared__ float sq0[BLK], sq1[BLK];
  __shared__ int swp[8], swn[8];
#if HAVE_ASYNC_LDS
  __shared__ float sbuf[BLK * NDIM];
#endif

  const float* base = pred + ((size_t)img * N_PRED + (size_t)chunk * BLK) * NDIM;

#if HAVE_ASYNC_LDS
  {
    // Each wave asynchronously stages its own 32 predictions (288 dwords)
    // into its private LDS region; only its own ASYNCcnt wait is needed.
    float* gsrc = const_cast<float*>(base) + w * (32 * NDIM);
    #pragma unroll
    for (int k = 0; k < NDIM; ++k) {
      __builtin_amdgcn_global_load_async_to_lds_b32(
          (as1_int_ptr)(gsrc + k * 32 + lane),
          (as3_int_ptr)(&sbuf[w * (32 * NDIM) + k * 32 + lane]),
          0, 0);
    }
  }
#endif

  if (tid < NT) {
    const float* t = tgt + ((size_t)img * NT + tid) * 5;
    float x = t[1], y = t[2], ww = t[3], hh = t[4];
    st1x[tid] = x - 0.5f * ww; st1y[tid] = y - 0.5f * hh;
    st2x[tid] = x + 0.5f * ww; st2y[tid] = y + 0.5f * hh;
    sta[tid]  = ww * hh;
  }
  __syncthreads();

  float p[NDIM];
#if HAVE_ASYNC_LDS
 #if HAVE_WAIT_ASYNC
  __builtin_amdgcn_s_wait_asynccnt(0);
 #else
  asm volatile("s_wait_asynccnt 0" ::: "memory");
 #endif
  asm volatile("" ::: "memory");
  #pragma unroll
  for (int j = 0; j < NDIM; ++j) p[j] = sbuf[tid * NDIM + j];
#else
  {
    const float* g = base + (size_t)tid * NDIM;
    #pragma unroll
    for (int j = 0; j < NDIM; ++j) p[j] = g[j];
  }
#endif

  const float px = p[0], py = p[1], pw = p[2], ph = p[3];
  const float p1x = px - 0.5f * pw, p1y = py - 0.5f * ph;
  const float p2x = px + 0.5f * pw, p2y = py + 0.5f * ph;
  const float pae = pw * ph + 1e-6f;

  // max_t IoU > 0.5 / < 0.3 without any division:
  // U = pa+ta-I+eps ; I>=0. iou>0.5 <=> U>0 && 2I>U ;
  // iou<0.3 <=> (U>0 && 10I<3U) || U<0  (U==0 -> inf/NaN -> neither).
  bool pos = false, neg = true;
  #pragma unroll 8
  for (int t = 0; t < NT; ++t) {
    float lox = fmaxf(p1x, st1x[t]);
    float loy = fmaxf(p1y, st1y[t]);
    float hix = fminf(p2x, st2x[t]);
    float hiy = fminf(p2y, st2y[t]);
    float dx = fmaxf(hix - lox, 0.f);
    float dy = fmaxf(hiy - loy, 0.f);
    float I  = dx * dy;
    float U  = pae + sta[t] - I;
    bool  up = U > 0.f;
    pos = pos || (up && (2.f * I > U));
    neg = neg && (up ? (10.f * I < 3.f * U) : (U < 0.f));
  }

  unsigned pm = ballot32(pos);
  unsigned nm = ballot32(neg);
  if (lane == 0) {
    posMask[(size_t)gid * 8 + w] = pm;
    swp[w] = __popc(pm);
    swn[w] = __popc(nm);
  }

  sq0[tid] = pos ? softplusf(-p[4]) : 0.f;
  sq1[tid] = neg ? softplusf(p[4]) : 0.f;
  __syncthreads();

  if (w == 0) {
    float s0 = reduce256_wave0(sq0, lane);
    float s1 = reduce256_wave0(sq1, lane);
    if (lane == 0) {
      objPart[(size_t)gid * 2 + 0] = s0;
      objPart[(size_t)gid * 2 + 1] = s1;
      int cp = 0, cn = 0;
      #pragma unroll
      for (int i = 0; i < 8; ++i) { cp += swp[i]; cn += swn[i]; }
      chunkPos[gid] = cp;
      chunkNeg[gid] = cn;
    }
  }
}

// ------------------------------------------------------------------
// K2: deterministic per-image exclusive scan of 300 chunk counts.
// ------------------------------------------------------------------
__global__ void k2_scan(const int* __restrict__ chunkPos,
                        const int* __restrict__ chunkNeg,
                        int* __restrict__ chunkOff,
                        int* __restrict__ nposImg,
                        int* __restrict__ nnegImg) {
  int i = threadIdx.x;
  if (i < NB) {
    int run = 0, nn = 0;
    for (int c = 0; c < CPB; ++c) {
      int g = i * CPB + c;
      chunkOff[g] = run;
      run += chunkPos[g];
      nn  += chunkNeg[g];
    }
    nposImg[i] = run;
    nnegImg[i] = nn;
  }
}

// ------------------------------------------------------------------
// K3: rank-dependent box-MSE and cross-entropy partials for positives.
// rank = image-global exclusive positive count (cumsum(pos)-1 at a positive).
// ------------------------------------------------------------------
__global__ __launch_bounds__(BLK) void k3_boxcls(const float* __restrict__ pred,
                                                 const float* __restrict__ tgt,
                                                 const unsigned* __restrict__ posMask,
                                                 const int* __restrict__ chunkOff,
                                                 float* __restrict__ bcPart) {
  const int gid  = blockIdx.x;
  const int img  = gid / CPB;
  const int chunk = gid - img * CPB;
  const int tid  = threadIdx.x;
  const int lane = tid & 31;
  const int w    = tid >> 5;

  __shared__ float stx[NT], sty[NT];
  __shared__ int stc[NT];
  __shared__ unsigned smask[8];
  __shared__ float sq0[BLK], sq1[BLK];

  if (tid < NT) {
    const float* t = tgt + ((size_t)img * NT + tid) * 5;
    stc[tid] = (int)t[0];
    stx[tid] = t[1];
    sty[tid] = t[2];
  }
  if (tid < 8) smask[tid] = posMask[(size_t)gid * 8 + tid];

  const float* g = pred + ((size_t)img * N_PRED + (size_t)chunk * BLK + tid) * NDIM;
  __builtin_prefetch(g, 0, 1);  // -> global_prefetch_b8 (L2-resident second pass)

  __syncthreads();

  unsigned m = smask[w];
  bool pos = (m >> lane) & 1u;
  int pre = __popc(m & ((1u << lane) - 1u));
  #pragma unroll
  for (int i = 0; i < 8; ++i)
    if (i < w) pre += __popc(smask[i]);

  float bv = 0.f, cv = 0.f;
  if (pos) {
    int rank = chunkOff[gid] + pre;       // exclusive count of positives before me
    int t = rank & (NT - 1);              // tidx = rank % 32
    float px = g[0], py = g[1];
    float c0 = g[5], c1 = g[6], c2 = g[7], c3 = g[8];
    float dx = px - stx[t], dy = py - sty[t];
    bv = dx * dx + dy * dy;
    float mx = fmaxf(fmaxf(c0, c1), fmaxf(c2, c3));
    float lse = mx + logf(expf(c0 - mx) + expf(c1 - mx) + expf(c2 - mx) + expf(c3 - mx));
    int tc = stc[t];
    float sel = tc == 0 ? c0 : (tc == 1 ? c1 : (tc == 2 ? c2 : c3));
    cv = lse - sel;
  }
  sq0[tid] = bv;
  sq1[tid] = cv;
  __syncthreads();

  if (w == 0) {
    float s0 = reduce256_wave0(sq0, lane);
    float s1 = reduce256_wave0(sq1, lane);
    if (lane == 0) {
      bcPart[(size_t)gid * 2 + 0] = s0;
      bcPart[(size_t)gid * 2 + 1] = s1;
    }
  }
}

// ------------------------------------------------------------------
// K4: fixed-order final combine -> (total, box, cls, obj).
// ------------------------------------------------------------------
__global__ void k4_final(const float* __restrict__ objPart,
                         const float* __restrict__ bcPart,
                         const int* __restrict__ nposImg,
                         const int* __restrict__ nnegImg,
                         float* __restrict__ out) {
  __shared__ float simg[NB][3];
  int i = threadIdx.x;
  if (i < NB) {
    float box = 0.f, cls = 0.f, op = 0.f, on = 0.f;
    for (int c = 0; c < CPB; ++c) {
      int g2 = i * CPB + c;
      op  += objPart[2 * g2 + 0];
      on  += objPart[2 * g2 + 1];
      box += bcPart[2 * g2 + 0];
      cls += bcPart[2 * g2 + 1];
    }
    float np = (float)nposImg[i];
    float nn = (float)nnegImg[i];
    simg[i][0] = np > 0.f ? box / (2.f * np) : 0.f;
    simg[i][1] = np > 0.f ? cls / np : 0.f;
    simg[i][2] = (np > 0.f ? op / np : 0.f) + (nn > 0.f ? on / nn : 0.f);
  }
  __syncthreads();
  if (i == 0) {
    float b = 0.f, c = 0.f, o = 0.f;
    for (int j = 0; j < NB; ++j) { b += simg[j][0]; c += simg[j][1]; o += simg[j][2]; }
    b /= (float)NB; c /= (float)NB; o /= (float)NB;
    out[0] = 0.05f * b + 0.5f * c + 1.0f * o;
    out[1] = b;
    out[2] = c;
    out[3] = o;
  }
}

extern "C" void kernel_launch(void* const* d_in, const int* in_sizes, int n_in,
                              void* d_out, int out_size, void* d_ws, size_t ws_size,
                              hipStream_t stream) {
  (void)in_sizes; (void)n_in; (void)out_size; (void)ws_size;
  const float* pred = (const float*)d_in[0];
  const float* tgt  = (const float*)d_in[1];

  char* p = (char*)d_ws;
  int* chunkPos = (int*)p;          p += (size_t)NCH * sizeof(int);
  int* chunkNeg = (int*)p;          p += (size_t)NCH * sizeof(int);
  int* chunkOff = (int*)p;          p += (size_t)NCH * sizeof(int);
  int* nposImg  = (int*)p;          p += (size_t)NB * sizeof(int);
  int* nnegImg  = (int*)p;          p += (size_t)NB * sizeof(int);
  unsigned* posMask = (unsigned*)p; p += (size_t)NCH * 8 * sizeof(unsigned);
  float* objPart = (float*)p;       p += (size_t)NCH * 2 * sizeof(float);
  float* bcPart  = (float*)p;       p += (size_t)NCH * 2 * sizeof(float);

  k1_iou  <<<NCH, BLK, 0, stream>>>(pred, tgt, chunkPos, chunkNeg, posMask, objPart);
  k2_scan <<<1, 32, 0, stream>>>(chunkPos, chunkNeg, chunkOff, nposImg, nnegImg);
  k3_boxcls<<<NCH, BLK, 0, stream>>>(pred, tgt, posMask, chunkOff, bcPart);
  k4_final<<<1, 32, 0, stream>>>(objPart, bcPart, nposImg, nnegImg, (float*)d_out);
}